// GraphConvolution_18545668784543
// MI455X (gfx1250) — compile-verified
//
#include <hip/hip_runtime.h>
#include <hip/hip_bf16.h>
#include <math.h>

// ---------------------------------------------------------------------------
// GCN layer on MI455X (gfx1250, wave32):
//   support = X @ W                      -> fp32 WMMA 16x16x4 (exact precision)
//   accum   = scatter-add over edges     -> L2-resident gather + native f32 atomics
//   out     = elu(accum + bias)          -> fused epilogue
// ---------------------------------------------------------------------------

typedef float v2f __attribute__((ext_vector_type(2)));
typedef float v8f __attribute__((ext_vector_type(8)));

#define D_IN  128
#define D_OUT 64

// ---- Kernel 0: zero the accumulator (deterministic per call) ----
__global__ __launch_bounds__(256) void zero_accum(float4* __restrict__ p, int n4) {
  int i = blockIdx.x * blockDim.x + threadIdx.x;
  if (i < n4) p[i] = make_float4(0.f, 0.f, 0.f, 0.f);
}

// ---- Kernel 1: support = X @ W via V_WMMA_F32_16X16X4_F32 ----
// One wave32 -> 16 rows x 64 cols tile. 8 waves/block -> 128 rows/block.
// A (16x4 f32): lanes 0-15 hold M=lane, K={0,1}; lanes 16-31 hold M=lane-16, K={2,3}.
// B (4x16 f32): lanes 0-15 hold N=lane, K={0,1}; lanes 16-31 hold N=lane-16, K={2,3}.
// D (16x16 f32): lane -> N = lane%16; VGPR v -> M = v + 8*(lane/16).
__global__ __launch_bounds__(256) void gemm_xw_wmma(
    const float* __restrict__ X, const float* __restrict__ W,
    float* __restrict__ support, int nRows)
{
  const int lane    = threadIdx.x & 31;
  const int wave    = threadIdx.x >> 5;
  const int rowBase = (blockIdx.x * 8 + wave) * 16;
  if (rowBase >= nRows) return;          // wave-uniform: EXEC stays all-ones

  const int half = lane >> 4;            // 0 = lanes 0-15, 1 = lanes 16-31
  const int lid  = lane & 15;
  const int kOff = half * 2;             // A/B K sub-offset per half-wave

  const float* __restrict__ xrow = X + (size_t)(rowBase + lid) * D_IN;

  v8f acc0 = {}, acc1 = {}, acc2 = {}, acc3 = {};

  #pragma unroll 4
  for (int kk = 0; kk < D_IN; kk += 4) {
    const int k0 = kk + kOff;
    // A fragment: two consecutive K values per lane -> one b64 load
    v2f a;
    a.x = xrow[k0];
    a.y = xrow[k0 + 1];
    // B fragments for the 4 column tiles (W is 32KB, L1/L2 resident)
    const float* __restrict__ w0 = W + (size_t)k0 * D_OUT + lid;
    const float* __restrict__ w1 = w0 + D_OUT;
    v2f b0, b1, b2, b3;
    b0.x = w0[0];  b0.y = w1[0];
    b1.x = w0[16]; b1.y = w1[16];
    b2.x = w0[32]; b2.y = w1[32];
    b3.x = w0[48]; b3.y = w1[48];
    acc0 = __builtin_amdgcn_wmma_f32_16x16x4_f32(false, a, false, b0, (short)0, acc0, false, false);
    acc1 = __builtin_amdgcn_wmma_f32_16x16x4_f32(false, a, false, b1, (short)0, acc1, false, false);
    acc2 = __builtin_amdgcn_wmma_f32_16x16x4_f32(false, a, false, b2, (short)0, acc2, false, false);
    acc3 = __builtin_amdgcn_wmma_f32_16x16x4_f32(false, a, false, b3, (short)0, acc3, false, false);
  }

  // Write back D tiles per ISA C/D layout.
  #pragma unroll
  for (int v = 0; v < 8; ++v) {
    const int r = rowBase + v + 8 * half;
    float* __restrict__ orow = support + (size_t)r * D_OUT + lid;
    orow[0]  = acc0[v];
    orow[16] = acc1[v];
    orow[32] = acc2[v];
    orow[48] = acc3[v];
  }
}

// ---- Kernel 2: edge scatter: accum[dst] += support[src] * val ----
// 16 lanes per edge, float4 per lane: GLOBAL_LOAD_B128 gather (L2-resident)
// + 4 native GLOBAL_ATOMIC_ADD_F32 (no return -> STOREcnt, fire-and-forget).
__global__ __launch_bounds__(256) void edge_scatter(
    const float* __restrict__ support,
    const int* __restrict__ edge_src, const int* __restrict__ edge_dst,
    const float* __restrict__ edge_vals,
    float* __restrict__ accum, int nEdges)
{
  const int t    = blockIdx.x * blockDim.x + threadIdx.x;
  const int e    = t >> 4;
  const int lane = t & 15;
  if (e >= nEdges) return;

  const int   s = edge_src[e];
  const int   d = edge_dst[e];
  const float v = edge_vals[e];

  const float4 m = *(const float4*)(support + (size_t)s * D_OUT + lane * 4);
  float* __restrict__ o = accum + (size_t)d * D_OUT + lane * 4;
  unsafeAtomicAdd(o + 0, m.x * v);
  unsafeAtomicAdd(o + 1, m.y * v);
  unsafeAtomicAdd(o + 2, m.z * v);
  unsafeAtomicAdd(o + 3, m.w * v);
}

// ---- Kernel 3: out = elu(accum + bias) ----
__global__ __launch_bounds__(256) void bias_elu(
    const float* __restrict__ accum, const float* __restrict__ bias,
    float* __restrict__ out, int n)
{
  int i = blockIdx.x * blockDim.x + threadIdx.x;
  if (i >= n) return;
  const float x = accum[i] + bias[i & (D_OUT - 1)];
  out[i] = (x > 0.f) ? x : expm1f(x);
}

extern "C" void kernel_launch(void* const* d_in, const int* in_sizes, int n_in,
                              void* d_out, int out_size, void* d_ws, size_t ws_size,
                              hipStream_t stream) {
  const float* X    = (const float*)d_in[0];
  const int*   esrc = (const int*)  d_in[1];
  const int*   edst = (const int*)  d_in[2];
  const float* eval = (const float*)d_in[3];
  const float* W    = (const float*)d_in[4];
  const float* bias = (const float*)d_in[5];
  float*       out  = (float*)d_out;

  const int nRows  = in_sizes[0] / D_IN;   // 50000
  const int nEdges = in_sizes[1];          // 800000
  const int nOut   = nRows * D_OUT;        // 3.2M floats

  float* support = (float*)d_ws;           // 12.8 MB
  float* accum   = support + (size_t)nOut; // 12.8 MB

  // 0) zero accumulator (float4 stores)
  {
    const int n4 = nOut / 4;
    zero_accum<<<(n4 + 255) / 256, 256, 0, stream>>>((float4*)accum, n4);
  }
  // 1) support = X @ W  (fp32 WMMA; 128 rows per 256-thread block)
  {
    const int rowTiles = (nRows + 15) / 16;
    const int blocks   = (rowTiles + 7) / 8;
    gemm_xw_wmma<<<blocks, 256, 0, stream>>>(X, W, support, nRows);
  }
  // 2) edge scatter with native f32 atomics (16 lanes/edge)
  {
    const long long lanes = (long long)nEdges * 16;
    const int blocks = (int)((lanes + 255) / 256);
    edge_scatter<<<blocks, 256, 0, stream>>>(support, esrc, edst, eval, accum, nEdges);
  }
  // 3) fused bias + ELU epilogue
  {
    bias_elu<<<(nOut + 255) / 256, 256, 0, stream>>>(accum, bias, out, nOut);
  }
}